// SimplePonderNet_48292612276446
// MI455X (gfx1250) — compile-verified
//
#include <hip/hip_runtime.h>
#include <hip/hip_bf16.h>
#include <math.h>

#define B_  8192
#define D_  2048
#define H_  1024
#define C_  1000
#define T_  10
#define HH_ 32

typedef float v2f __attribute__((ext_vector_type(2)));
typedef float v4f __attribute__((ext_vector_type(4)));
typedef float v8f __attribute__((ext_vector_type(8)));

__device__ __forceinline__ v8f wmma_f32(v2f a, v2f b, v8f c) {
  // D = A(16x4 f32) x B(4x16 f32) + C(16x16 f32), exact fp32 matrix pipe
  return __builtin_amdgcn_wmma_f32_16x16x4_f32(false, a, false, b, (short)0, c,
                                               false, false);
}
__device__ __forceinline__ v2f vlo(v4f v) { return __builtin_shufflevector(v, v, 0, 1); }
__device__ __forceinline__ v2f vhi(v4f v) { return __builtin_shufflevector(v, v, 2, 3); }

__device__ __forceinline__ float sigmoidf_(float x) {
  return 1.0f / (1.0f + expf(-x));
}

// ---------------------------------------------------------------------------
// Generic GEMM: Cout[M,N] = A[M,K] @ W[N,K]^T + bias[N]
// wave tile: 16 rows x 64 cols (4 WMMA chains), K-chunk = 8 via permuted-K
// b128 fragment loads + register double-buffering.
// grid: (M/128, N/64), 256 threads.
// ---------------------------------------------------------------------------
__global__ __launch_bounds__(256) void ponder_gemm_bias(
    const float* __restrict__ A, const float* __restrict__ W,
    const float* __restrict__ bias, float* __restrict__ Cout,
    int M, int N, int K) {
  const int lane = threadIdx.x & 31;
  const int wave = threadIdx.x >> 5;
  const int row0 = (blockIdx.x * 8 + wave) * 16;
  const int col0 = blockIdx.y * 64;
  const int m  = lane & 15;
  const int kh = lane >> 4;   // lane-half: global k offset 0 / +4 within chunk
  const int n  = lane & 15;

  const float* ap  = A + (size_t)(row0 + m) * K + 4 * kh;
  const float* wp0 = W + (size_t)(col0 + n)      * K + 4 * kh;
  const float* wp1 = W + (size_t)(col0 + 16 + n) * K + 4 * kh;
  const float* wp2 = W + (size_t)(col0 + 32 + n) * K + 4 * kh;
  const float* wp3 = W + (size_t)(col0 + 48 + n) * K + 4 * kh;

  v8f acc0 = {}, acc1 = {}, acc2 = {}, acc3 = {};
  v4f ac  = *(const v4f*)(ap);
  v4f b0c = *(const v4f*)(wp0);
  v4f b1c = *(const v4f*)(wp1);
  v4f b2c = *(const v4f*)(wp2);
  v4f b3c = *(const v4f*)(wp3);
#pragma unroll 1
  for (int k = 8; k < K; k += 8) {
    v4f an  = *(const v4f*)(ap + k);     // prefetch next chunk
    v4f b0n = *(const v4f*)(wp0 + k);
    v4f b1n = *(const v4f*)(wp1 + k);
    v4f b2n = *(const v4f*)(wp2 + k);
    v4f b3n = *(const v4f*)(wp3 + k);
    acc0 = wmma_f32(vlo(ac), vlo(b0c), acc0);
    acc1 = wmma_f32(vlo(ac), vlo(b1c), acc1);
    acc2 = wmma_f32(vlo(ac), vlo(b2c), acc2);
    acc3 = wmma_f32(vlo(ac), vlo(b3c), acc3);
    acc0 = wmma_f32(vhi(ac), vhi(b0c), acc0);
    acc1 = wmma_f32(vhi(ac), vhi(b1c), acc1);
    acc2 = wmma_f32(vhi(ac), vhi(b2c), acc2);
    acc3 = wmma_f32(vhi(ac), vhi(b3c), acc3);
    ac = an; b0c = b0n; b1c = b1n; b2c = b2n; b3c = b3n;
  }
  acc0 = wmma_f32(vlo(ac), vlo(b0c), acc0);
  acc1 = wmma_f32(vlo(ac), vlo(b1c), acc1);
  acc2 = wmma_f32(vlo(ac), vlo(b2c), acc2);
  acc3 = wmma_f32(vlo(ac), vlo(b3c), acc3);
  acc0 = wmma_f32(vhi(ac), vhi(b0c), acc0);
  acc1 = wmma_f32(vhi(ac), vhi(b1c), acc1);
  acc2 = wmma_f32(vhi(ac), vhi(b2c), acc2);
  acc3 = wmma_f32(vhi(ac), vhi(b3c), acc3);

#pragma unroll
  for (int i = 0; i < 8; ++i) {
    const int row = row0 + i + 8 * kh;
    float* orow = Cout + (size_t)row * N;
    int c0 = col0 + n;
    if (c0 < N)      orow[c0]      = acc0[i] + bias[c0];
    if (c0+16 < N)   orow[c0 + 16] = acc1[i] + bias[c0 + 16];
    if (c0+32 < N)   orow[c0 + 32] = acc2[i] + bias[c0 + 32];
    if (c0+48 < N)   orow[c0 + 48] = acc3[i] + bias[c0 + 48];
  }
}

// ---------------------------------------------------------------------------
// LayerNorm + exact GELU, one block per row of H_.
// ---------------------------------------------------------------------------
__global__ __launch_bounds__(256) void ponder_ln_gelu(
    const float* __restrict__ z, const float* __restrict__ g,
    const float* __restrict__ b, float* __restrict__ out) {
  __shared__ float red[256];
  const int row = blockIdx.x, tid = threadIdx.x;
  const float* zr = z + (size_t)row * H_;
  float s = 0.f, s2 = 0.f;
  for (int i = tid; i < H_; i += 256) { float v = zr[i]; s += v; s2 += v * v; }
  red[tid] = s; __syncthreads();
  for (int o = 128; o > 0; o >>= 1) { if (tid < o) red[tid] += red[tid + o]; __syncthreads(); }
  const float mu = red[0] * (1.0f / H_);
  __syncthreads();
  red[tid] = s2; __syncthreads();
  for (int o = 128; o > 0; o >>= 1) { if (tid < o) red[tid] += red[tid + o]; __syncthreads(); }
  const float var = red[0] * (1.0f / H_) - mu * mu;
  const float inv = rsqrtf(var + 1e-5f);
  float* orow = out + (size_t)row * H_;
  for (int i = tid; i < H_; i += 256) {
    float v = (zr[i] - mu) * inv * g[i] + b[i];
    orow[i] = 0.5f * v * (1.0f + erff(v * 0.70710678118654752f));
  }
}

// ---------------------------------------------------------------------------
// Fused GRU step: one wave -> 16x16 tile of h_new, SIX WMMA chains (no gi/gh
// scratch tensors). K-chunk 8, b128 loads, double-buffered.
// grid: (B_/128, H_/16), 256 threads.
// ---------------------------------------------------------------------------
__global__ __launch_bounds__(256) void ponder_gru_step(
    const float* __restrict__ h_in, const float* __restrict__ W_ih,
    const float* __restrict__ W_hh, const float* __restrict__ b_ih,
    const float* __restrict__ b_hh, float* __restrict__ h_out) {
  const int lane = threadIdx.x & 31;
  const int wave = threadIdx.x >> 5;
  const int row0 = (blockIdx.x * 8 + wave) * 16;
  const int col0 = blockIdx.y * 16;
  const int m  = lane & 15;
  const int kh = lane >> 4;
  const int n  = lane & 15;

  const float* ap  = h_in + (size_t)(row0 + m) * H_ + 4 * kh;
  const size_t wro = (size_t)(col0 + n) * H_ + 4 * kh;
  const float* ir  = W_ih + wro;
  const float* iz  = W_ih + (size_t)H_ * H_ + wro;
  const float* in_ = W_ih + (size_t)2 * H_ * H_ + wro;
  const float* hr  = W_hh + wro;
  const float* hz  = W_hh + (size_t)H_ * H_ + wro;
  const float* hn  = W_hh + (size_t)2 * H_ * H_ + wro;

  v8f a_ir = {}, a_iz = {}, a_in = {}, a_hr = {}, a_hz = {}, a_hn = {};
  v4f ac   = *(const v4f*)(ap);
  v4f birc = *(const v4f*)(ir);
  v4f bizc = *(const v4f*)(iz);
  v4f binc = *(const v4f*)(in_);
  v4f bhrc = *(const v4f*)(hr);
  v4f bhzc = *(const v4f*)(hz);
  v4f bhnc = *(const v4f*)(hn);
#pragma unroll 1
  for (int k = 8; k < H_; k += 8) {
    v4f an   = *(const v4f*)(ap + k);
    v4f birn = *(const v4f*)(ir + k);
    v4f bizn = *(const v4f*)(iz + k);
    v4f binn = *(const v4f*)(in_ + k);
    v4f bhrn = *(const v4f*)(hr + k);
    v4f bhzn = *(const v4f*)(hz + k);
    v4f bhnn = *(const v4f*)(hn + k);
    a_ir = wmma_f32(vlo(ac), vlo(birc), a_ir);
    a_iz = wmma_f32(vlo(ac), vlo(bizc), a_iz);
    a_in = wmma_f32(vlo(ac), vlo(binc), a_in);
    a_hr = wmma_f32(vlo(ac), vlo(bhrc), a_hr);
    a_hz = wmma_f32(vlo(ac), vlo(bhzc), a_hz);
    a_hn = wmma_f32(vlo(ac), vlo(bhnc), a_hn);
    a_ir = wmma_f32(vhi(ac), vhi(birc), a_ir);
    a_iz = wmma_f32(vhi(ac), vhi(bizc), a_iz);
    a_in = wmma_f32(vhi(ac), vhi(binc), a_in);
    a_hr = wmma_f32(vhi(ac), vhi(bhrc), a_hr);
    a_hz = wmma_f32(vhi(ac), vhi(bhzc), a_hz);
    a_hn = wmma_f32(vhi(ac), vhi(bhnc), a_hn);
    ac = an; birc = birn; bizc = bizn; binc = binn;
    bhrc = bhrn; bhzc = bhzn; bhnc = bhnn;
  }
  a_ir = wmma_f32(vlo(ac), vlo(birc), a_ir);
  a_iz = wmma_f32(vlo(ac), vlo(bizc), a_iz);
  a_in = wmma_f32(vlo(ac), vlo(binc), a_in);
  a_hr = wmma_f32(vlo(ac), vlo(bhrc), a_hr);
  a_hz = wmma_f32(vlo(ac), vlo(bhzc), a_hz);
  a_hn = wmma_f32(vlo(ac), vlo(bhnc), a_hn);
  a_ir = wmma_f32(vhi(ac), vhi(birc), a_ir);
  a_iz = wmma_f32(vhi(ac), vhi(bizc), a_iz);
  a_in = wmma_f32(vhi(ac), vhi(binc), a_in);
  a_hr = wmma_f32(vhi(ac), vhi(bhrc), a_hr);
  a_hz = wmma_f32(vhi(ac), vhi(bhzc), a_hz);
  a_hn = wmma_f32(vhi(ac), vhi(bhnc), a_hn);

  const int col = col0 + n;
  const float cir = b_ih[col], ciz = b_ih[H_ + col], cin = b_ih[2 * H_ + col];
  const float chr = b_hh[col], chz = b_hh[H_ + col], chn = b_hh[2 * H_ + col];
#pragma unroll
  for (int i = 0; i < 8; ++i) {
    const int row = row0 + i + 8 * kh;
    const float hold = h_in[(size_t)row * H_ + col];
    const float r  = sigmoidf_((a_ir[i] + cir) + (a_hr[i] + chr));
    const float zt = sigmoidf_((a_iz[i] + ciz) + (a_hz[i] + chz));
    const float nn = tanhf((a_in[i] + cin) + r * (a_hn[i] + chn));
    h_out[(size_t)row * H_ + col] = (1.0f - zt) * nn + zt * hold;
  }
}

// ---------------------------------------------------------------------------
// Halt head + online PonderNet recurrence. HALT_H==32==wave32: lane o owns
// output o. One wave per batch row; grid B_/8, 256 threads.
// ---------------------------------------------------------------------------
__global__ __launch_bounds__(256) void ponder_halt_step(
    const float* __restrict__ h, const float* __restrict__ h1_W,
    const float* __restrict__ h1_b, const float* __restrict__ h2_W,
    const float* __restrict__ h2_b, float* __restrict__ p_buf,
    float* __restrict__ surv, float* __restrict__ psum,
    float* __restrict__ out_hdist, float* __restrict__ out_esteps, int t) {
  const int lane = threadIdx.x & 31;
  const int b = blockIdx.x * 8 + (threadIdx.x >> 5);
  const float* hrow = h + (size_t)b * H_;
  const float* wrow = h1_W + (size_t)lane * H_;
  float u = h1_b[lane];
  for (int k = 0; k < H_; k += 4) {
    float4 hv = *(const float4*)(hrow + k);
    float4 wv = *(const float4*)(wrow + k);
    u += hv.x * wv.x + hv.y * wv.y + hv.z * wv.z + hv.w * wv.w;
  }
  float v = fmaxf(u, 0.0f) * h2_W[lane];
#pragma unroll
  for (int msk = 16; msk >= 1; msk >>= 1) v += __shfl_xor(v, msk, 32);
  if (lane == 0) {
    const float halt = sigmoidf_(v + h2_b[0]);
    const float s  = (t == 1) ? 1.0f : surv[b];
    const float ps = (t == 1) ? 0.0f : psum[b];
    float pv = halt * s;
    if (t == T_) {                           // fold the remainder into p_T
      const float rem = 1.0f - (ps + pv);
      pv += fmaxf(rem, 0.0f);
    }
    surv[b] = s * (1.0f - halt + 1e-8f);
    psum[b] = ps + pv;
    p_buf[b] = pv;
    out_hdist[(size_t)b * T_ + (t - 1)] = pv;
    const float e = pv * (float)t;
    out_esteps[b] = (t == 1) ? e : (out_esteps[b] + e);
  }
}

// ---------------------------------------------------------------------------
// Classifier GEMM with online halt-weighted accumulation into d_out.
// out[b,c] (= / +=) p[b] * (h[b]@cls_W[c] + cls_b[c]).  16x64 per wave.
// B-operand lane n only contributes to output column n, and cols >= C_ are
// never stored, so clamped (safe) loads need no masking.
// ---------------------------------------------------------------------------
__global__ __launch_bounds__(256) void ponder_cls_step(
    const float* __restrict__ h, const float* __restrict__ W,
    const float* __restrict__ bias, const float* __restrict__ p,
    float* __restrict__ out, int first) {
  const int lane = threadIdx.x & 31;
  const int wave = threadIdx.x >> 5;
  const int row0 = (blockIdx.x * 8 + wave) * 16;
  const int col0 = blockIdx.y * 64;
  const int m  = lane & 15;
  const int kh = lane >> 5 ? 0 : (lane >> 4);  // == lane>>4 (0/1)
  const int n  = lane & 15;
  const int kh_ = lane >> 4;

  const float* ap = h + (size_t)(row0 + m) * H_ + 4 * kh_;
  int nr0 = col0 + n,      nr1 = col0 + 16 + n;
  int nr2 = col0 + 32 + n, nr3 = col0 + 48 + n;
  nr0 = (nr0 < C_) ? nr0 : C_ - 1;   // clamp for safe loads only
  nr1 = (nr1 < C_) ? nr1 : C_ - 1;
  nr2 = (nr2 < C_) ? nr2 : C_ - 1;
  nr3 = (nr3 < C_) ? nr3 : C_ - 1;
  const float* wp0 = W + (size_t)nr0 * H_ + 4 * kh_;
  const float* wp1 = W + (size_t)nr1 * H_ + 4 * kh_;
  const float* wp2 = W + (size_t)nr2 * H_ + 4 * kh_;
  const float* wp3 = W + (size_t)nr3 * H_ + 4 * kh_;
  (void)kh;

  v8f acc0 = {}, acc1 = {}, acc2 = {}, acc3 = {};
  v4f ac  = *(const v4f*)(ap);
  v4f b0c = *(const v4f*)(wp0);
  v4f b1c = *(const v4f*)(wp1);
  v4f b2c = *(const v4f*)(wp2);
  v4f b3c = *(const v4f*)(wp3);
#pragma unroll 1
  for (int k = 8; k < H_; k += 8) {
    v4f an  = *(const v4f*)(ap + k);
    v4f b0n = *(const v4f*)(wp0 + k);
    v4f b1n = *(const v4f*)(wp1 + k);
    v4f b2n = *(const v4f*)(wp2 + k);
    v4f b3n = *(const v4f*)(wp3 + k);
    acc0 = wmma_f32(vlo(ac), vlo(b0c), acc0);
    acc1 = wmma_f32(vlo(ac), vlo(b1c), acc1);
    acc2 = wmma_f32(vlo(ac), vlo(b2c), acc2);
    acc3 = wmma_f32(vlo(ac), vlo(b3c), acc3);
    acc0 = wmma_f32(vhi(ac), vhi(b0c), acc0);
    acc1 = wmma_f32(vhi(ac), vhi(b1c), acc1);
    acc2 = wmma_f32(vhi(ac), vhi(b2c), acc2);
    acc3 = wmma_f32(vhi(ac), vhi(b3c), acc3);
    ac = an; b0c = b0n; b1c = b1n; b2c = b2n; b3c = b3n;
  }
  acc0 = wmma_f32(vlo(ac), vlo(b0c), acc0);
  acc1 = wmma_f32(vlo(ac), vlo(b1c), acc1);
  acc2 = wmma_f32(vlo(ac), vlo(b2c), acc2);
  acc3 = wmma_f32(vlo(ac), vlo(b3c), acc3);
  acc0 = wmma_f32(vhi(ac), vhi(b0c), acc0);
  acc1 = wmma_f32(vhi(ac), vhi(b1c), acc1);
  acc2 = wmma_f32(vhi(ac), vhi(b2c), acc2);
  acc3 = wmma_f32(vhi(ac), vhi(b3c), acc3);

#pragma unroll
  for (int i = 0; i < 8; ++i) {
    const int row = row0 + i + 8 * kh_;
    const float pr = p[row];
    float* orow = out + (size_t)row * C_;
    const int c0 = col0 + n;
#pragma unroll
    for (int tsub = 0; tsub < 4; ++tsub) {
      const int c = c0 + 16 * tsub;
      if (c < C_) {
        float a = (tsub == 0) ? acc0[i] : (tsub == 1) ? acc1[i]
                 : (tsub == 2) ? acc2[i] : acc3[i];
        const float val = pr * (a + bias[c]);
        orow[c] = first ? val : (orow[c] + val);
      }
    }
  }
}

// ---------------------------------------------------------------------------
// Softmax + confidence outputs. One block per row.  max prob == 1/sum(exp).
// ---------------------------------------------------------------------------
__global__ __launch_bounds__(256) void ponder_finalize(
    const float* __restrict__ wl, float* __restrict__ probs,
    float* __restrict__ conf, float* __restrict__ sconf,
    const float* __restrict__ esteps, float* __restrict__ cu) {
  __shared__ float red[256];
  const int b = blockIdx.x, tid = threadIdx.x;
  const float* row = wl + (size_t)b * C_;
  float mx = -INFINITY;
  for (int i = tid; i < C_; i += 256) mx = fmaxf(mx, row[i]);
  red[tid] = mx; __syncthreads();
  for (int o = 128; o > 0; o >>= 1) { if (tid < o) red[tid] = fmaxf(red[tid], red[tid + o]); __syncthreads(); }
  mx = red[0]; __syncthreads();
  float s = 0.f;
  for (int i = tid; i < C_; i += 256) s += expf(row[i] - mx);
  red[tid] = s; __syncthreads();
  for (int o = 128; o > 0; o >>= 1) { if (tid < o) red[tid] += red[tid + o]; __syncthreads(); }
  const float sum = red[0];
  const float inv = 1.0f / sum;
  float* prow = probs + (size_t)b * C_;
  for (int i = tid; i < C_; i += 256) prow[i] = expf(row[i] - mx) * inv;
  if (tid == 0) {
    const float maxp = inv;            // exp(mx-mx)/sum
    const float es = esteps[b];
    const float u = es * (1.0f / (float)T_);
    sconf[b] = maxp;
    cu[b] = u;
    conf[b] = 0.7f * maxp + 0.3f * (1.0f - u);
  }
}

// ---------------------------------------------------------------------------
extern "C" void kernel_launch(void* const* d_in, const int* in_sizes, int n_in,
                              void* d_out, int out_size, void* d_ws,
                              size_t ws_size, hipStream_t stream) {
  (void)in_sizes; (void)n_in; (void)out_size; (void)ws_size;
  const float* x     = (const float*)d_in[0];
  const float* enc_W = (const float*)d_in[1];
  const float* enc_b = (const float*)d_in[2];
  const float* ln_g  = (const float*)d_in[3];
  const float* ln_b  = (const float*)d_in[4];
  const float* W_ih  = (const float*)d_in[5];
  const float* W_hh  = (const float*)d_in[6];
  const float* b_ih  = (const float*)d_in[7];
  const float* b_hh  = (const float*)d_in[8];
  const float* cls_W = (const float*)d_in[9];
  const float* cls_b = (const float*)d_in[10];
  const float* h1_W  = (const float*)d_in[11];
  const float* h1_b  = (const float*)d_in[12];
  const float* h2_W  = (const float*)d_in[13];
  const float* h2_b  = (const float*)d_in[14];

  float* ws = (float*)d_ws;
  float* hA    = ws;                            // [B,H] ping
  float* hB    = ws + (size_t)B_ * H_;          // [B,H] pong (also encoder z)
  float* p_buf = ws + (size_t)2 * B_ * H_;      // [B]
  float* surv  = p_buf + B_;                    // [B]
  float* psum  = surv + B_;                     // [B]

  float* out_wl     = (float*)d_out;            // [B,C]
  float* out_probs  = out_wl + (size_t)B_ * C_; // [B,C]
  float* out_conf   = out_probs + (size_t)B_ * C_; // [B]
  float* out_sconf  = out_conf + B_;            // [B]
  float* out_esteps = out_sconf + B_;           // [B]
  float* out_hdist  = out_esteps + B_;          // [B,T]
  float* out_cu     = out_hdist + (size_t)B_ * T_; // [B]

  // encoder: z = x @ enc_W^T + enc_b   (M=B, N=H, K=D) -> hB
  ponder_gemm_bias<<<dim3(B_ / 128, H_ / 64), 256, 0, stream>>>(
      x, enc_W, enc_b, hB, B_, H_, D_);
  // h0 = gelu(layernorm(z)) -> hA
  ponder_ln_gelu<<<B_, 256, 0, stream>>>(hB, ln_g, ln_b, hA);

  float* hin = hA;
  float* hout = hB;
  for (int t = 1; t <= T_; ++t) {
    ponder_gru_step<<<dim3(B_ / 128, H_ / 16), 256, 0, stream>>>(
        hin, W_ih, W_hh, b_ih, b_hh, hout);
    ponder_halt_step<<<B_ / 8, 256, 0, stream>>>(
        hout, h1_W, h1_b, h2_W, h2_b, p_buf, surv, psum, out_hdist,
        out_esteps, t);
    ponder_cls_step<<<dim3(B_ / 128, 16), 256, 0, stream>>>(
        hout, cls_W, cls_b, p_buf, out_wl, (t == 1) ? 1 : 0);
    float* tmp = hout; hout = hin; hin = tmp;
  }
  ponder_finalize<<<B_, 256, 0, stream>>>(out_wl, out_probs, out_conf,
                                          out_sconf, out_esteps, out_cu);
}